// SSIM_27410481283584
// MI455X (gfx1250) — compile-verified
//
#include <hip/hip_runtime.h>

typedef __attribute__((ext_vector_type(2))) float v2f;
typedef __attribute__((ext_vector_type(8))) float v8f;

#define WS        11
#define TILE      16
#define HALO      5          // (WS-1)/2
#define IN_DIM    26         // TILE + WS - 1
#define STAGE_R   32         // padded rows (2 x 16-row WMMA tiles)
#define STAGE_C   28         // padded K (7 chunks of 4)
#define T_STRIDE  18         // LDS stride for intermediate, bank-conflict free
#define NCHUNK    7
#define IMG       512
#define WPB       2          // waves per block

struct WaveLds {
  float sx[STAGE_R * STAGE_C];       // staged img1 region (reflect + zero pad)
  float sy[STAGE_R * STAGE_C];       // staged img2 region
  float T[5][STAGE_R * T_STRIDE];    // horizontally-filtered x, y, xx, yy, xy
};

__device__ __forceinline__ int reflect512(int i) {
  i = (i < 0) ? -i : i;
  i = (i >= IMG) ? (2 * IMG - 2 - i) : i;
  return i;
}

__global__ __launch_bounds__(32 * WPB)
void ssim_wmma(const float* __restrict__ img1, const float* __restrict__ img2,
               const float* __restrict__ k2d, float* __restrict__ out) {
  __shared__ WaveLds lds[WPB];
  __shared__ float g[WS];

  const int tid  = threadIdx.x;
  const int wave = tid >> 5;
  const int lane = tid & 31;

  // Recover 1-D gaussian: rows of outer(g,g) sum to g[i] (since sum(g)==1).
  if (tid < WS) {
    float s = 0.f;
    #pragma unroll
    for (int j = 0; j < WS; ++j) s += k2d[tid * WS + j];
    g[tid] = s;
  }
  __syncthreads();

  const int tile = blockIdx.x * WPB + wave;   // grid sized exactly: 48*32*32 tiles
  const int bx = tile & 31;
  const int by = (tile >> 5) & 31;
  const int im = tile >> 10;
  const size_t base = (size_t)im * (IMG * IMG);

  WaveLds& L = lds[wave];

  // ---- Stage 26x26 reflect-padded region, zero-padded to 32x28 ----
  const int gr0 = by * TILE - HALO;
  const int gc0 = bx * TILE - HALO;
  for (int idx = lane; idx < STAGE_R * STAGE_C; idx += 32) {
    const int lr = idx / STAGE_C;
    const int lc = idx - lr * STAGE_C;
    float x = 0.f, y = 0.f;
    if (lr < IN_DIM && lc < IN_DIM) {
      const int r = reflect512(gr0 + lr);
      const int c = reflect512(gc0 + lc);
      x = img1[base + (size_t)r * IMG + c];
      y = img2[base + (size_t)r * IMG + c];
    }
    L.sx[idx] = x;
    L.sy[idx] = y;
  }
  __syncthreads();

  // ---- Per-lane band-weight chunks; identical value for pass-1 B and pass-2 A:
  // element value = g[K - (lane&15)] with K = 4*ch + (lane>>4)*2 + j
  const int n15   = lane & 15;
  const int khalf = (lane >> 4) * 2;
  v2f w[NCHUNK];
  #pragma unroll
  for (int ch = 0; ch < NCHUNK; ++ch) {
    const int k0 = 4 * ch + khalf;
    const int d0 = k0 - n15, d1 = k0 + 1 - n15;
    w[ch][0] = (d0 >= 0 && d0 < WS) ? g[d0] : 0.f;
    w[ch][1] = (d1 >= 0 && d1 < WS) ? g[d1] : 0.f;
  }

  // ---- Pass 1 (horizontal): T = In(32x28) x Wh(28x16), two 16-row tiles ----
  #pragma unroll
  for (int m = 0; m < 2; ++m) {
    v8f aX = {}, aY = {}, aXX = {}, aYY = {}, aXY = {};
    const int row = m * TILE + n15;                 // A-matrix M = lane&15
    #pragma unroll
    for (int ch = 0; ch < NCHUNK; ++ch) {
      const int k0 = 4 * ch + khalf;
      const float x0 = L.sx[row * STAGE_C + k0];
      const float x1 = L.sx[row * STAGE_C + k0 + 1];
      const float y0 = L.sy[row * STAGE_C + k0];
      const float y1 = L.sy[row * STAGE_C + k0 + 1];
      v2f ax  = {x0, x1},       ay  = {y0, y1};
      v2f axx = {x0*x0, x1*x1}, ayy = {y0*y0, y1*y1}, axy = {x0*y0, x1*y1};
      aX  = __builtin_amdgcn_wmma_f32_16x16x4_f32(false, ax,  false, w[ch], (short)0, aX,  false, false);
      aY  = __builtin_amdgcn_wmma_f32_16x16x4_f32(false, ay,  false, w[ch], (short)0, aY,  false, false);
      aXX = __builtin_amdgcn_wmma_f32_16x16x4_f32(false, axx, false, w[ch], (short)0, aXX, false, false);
      aYY = __builtin_amdgcn_wmma_f32_16x16x4_f32(false, ayy, false, w[ch], (short)0, aYY, false, false);
      aXY = __builtin_amdgcn_wmma_f32_16x16x4_f32(false, axy, false, w[ch], (short)0, aXY, false, false);
    }
    // C/D layout: VGPR v <-> row (lane>>4)*8 + v, col = lane&15
    const int tr0 = m * TILE + (lane >> 4) * 8;
    #pragma unroll
    for (int v = 0; v < 8; ++v) {
      const int tr = tr0 + v;
      L.T[0][tr * T_STRIDE + n15] = aX[v];
      L.T[1][tr * T_STRIDE + n15] = aY[v];
      L.T[2][tr * T_STRIDE + n15] = aXX[v];
      L.T[3][tr * T_STRIDE + n15] = aYY[v];
      L.T[4][tr * T_STRIDE + n15] = aXY[v];
    }
  }
  __syncthreads();

  // ---- Pass 2 (vertical): Out = Wv(16x28) x T(28x16) ----
  v8f oX = {}, oY = {}, oXX = {}, oYY = {}, oXY = {};
  #pragma unroll
  for (int ch = 0; ch < NCHUNK; ++ch) {
    const int r0 = 4 * ch + khalf;                  // B-matrix K rows for this lane half
    v2f b;
    b[0] = L.T[0][r0 * T_STRIDE + n15]; b[1] = L.T[0][(r0 + 1) * T_STRIDE + n15];
    oX  = __builtin_amdgcn_wmma_f32_16x16x4_f32(false, w[ch], false, b, (short)0, oX,  false, false);
    b[0] = L.T[1][r0 * T_STRIDE + n15]; b[1] = L.T[1][(r0 + 1) * T_STRIDE + n15];
    oY  = __builtin_amdgcn_wmma_f32_16x16x4_f32(false, w[ch], false, b, (short)0, oY,  false, false);
    b[0] = L.T[2][r0 * T_STRIDE + n15]; b[1] = L.T[2][(r0 + 1) * T_STRIDE + n15];
    oXX = __builtin_amdgcn_wmma_f32_16x16x4_f32(false, w[ch], false, b, (short)0, oXX, false, false);
    b[0] = L.T[3][r0 * T_STRIDE + n15]; b[1] = L.T[3][(r0 + 1) * T_STRIDE + n15];
    oYY = __builtin_amdgcn_wmma_f32_16x16x4_f32(false, w[ch], false, b, (short)0, oYY, false, false);
    b[0] = L.T[4][r0 * T_STRIDE + n15]; b[1] = L.T[4][(r0 + 1) * T_STRIDE + n15];
    oXY = __builtin_amdgcn_wmma_f32_16x16x4_f32(false, w[ch], false, b, (short)0, oXY, false, false);
  }

  // ---- SSIM pointwise + store ----
  const float C1 = 1e-4f, C2 = 9e-4f;
  const int orow0 = by * TILE + (lane >> 4) * 8;
  const int ocol  = bx * TILE + n15;
  #pragma unroll
  for (int v = 0; v < 8; ++v) {
    const float mu1 = oX[v], mu2 = oY[v];
    const float m11 = mu1 * mu1, m22 = mu2 * mu2, m12 = mu1 * mu2;
    const float s1  = oXX[v] - m11;
    const float s2  = oYY[v] - m22;
    const float s12 = oXY[v] - m12;
    const float num = (2.f * m12 + C1) * (2.f * s12 + C2);
    const float den = (m11 + m22 + C1) * (s1 + s2 + C2);
    out[base + (size_t)(orow0 + v) * IMG + ocol] = num / den;
  }
}

extern "C" void kernel_launch(void* const* d_in, const int* in_sizes, int n_in,
                              void* d_out, int out_size, void* d_ws, size_t ws_size,
                              hipStream_t stream) {
  const float* img1 = (const float*)d_in[0];
  const float* img2 = (const float*)d_in[1];
  const float* k2d  = (const float*)d_in[2];
  float* out = (float*)d_out;

  const int ntiles = 48 * 32 * 32;   // (B*C) x (512/16)^2 = 49152, exact
  dim3 grid(ntiles / WPB);
  dim3 block(32 * WPB);
  hipLaunchKernelGGL(ssim_wmma, grid, block, 0, stream, img1, img2, k2d, out);
}